// GSNN_48189533061452
// MI455X (gfx1250) — compile-verified
//
#include <hip/hip_runtime.h>
#include <hip/hip_bf16.h>

// Problem constants (from reference setup_inputs)
#define NN   2000      // nodes
#define NE   20000     // edges
#define CH   16        // channels
#define BATCH 128      // batch
#define LAYERS 4
#define EPSV 1e-5f

typedef __attribute__((ext_vector_type(2))) float v2f;
typedef __attribute__((ext_vector_type(8))) float v8f;

// ---------------------------------------------------------------------------
// Seed: x0[e][b] = xe[e][b] = x[b][src[e]];  w3m[e][c] = w3[e][c]*fm[src[e]]
// ---------------------------------------------------------------------------
__global__ __launch_bounds__(128) void k_seed(
    const float* __restrict__ x, const int* __restrict__ ei,
    const float* __restrict__ w3, const unsigned char* __restrict__ fmask,
    float* __restrict__ x0, float* __restrict__ xe, float* __restrict__ w3m)
{
    const int e = blockIdx.x;
    const int b = threadIdx.x;
    const int s = ei[e];                       // src
    const float v = x[(size_t)b * NN + s];
    x0[(size_t)e * BATCH + b] = v;
    xe[(size_t)e * BATCH + b] = v;
    if (b < CH) {
        const float fm = fmask[s] ? 1.0f : 0.0f;
        w3m[e * CH + b] = w3[e * CH + b] * fm;
    }
}

// ---------------------------------------------------------------------------
// Init h[n][b][c] = b1[n][c]
// ---------------------------------------------------------------------------
__global__ __launch_bounds__(256) void k_init_h(
    const float* __restrict__ b1, float* __restrict__ h)
{
    const int i = blockIdx.x * 256 + threadIdx.x;   // i = n*2048 + b*16 + c
    if (i < NN * BATCH * CH) {
        const int n = i >> 11;
        const int c = i & 15;
        h[i] = b1[n * CH + c];
    }
}

// ---------------------------------------------------------------------------
// Edge scatter: h[dst[e]][b][c] += xe[e][b] * w1[e][c]   (fp32 global atomics)
// ---------------------------------------------------------------------------
__global__ __launch_bounds__(256) void k_scatter(
    const float* __restrict__ xe, const int* __restrict__ ei,
    const float* __restrict__ w1, float* __restrict__ h)
{
    const int e = blockIdx.x;
    const int d = ei[NE + e];                  // dst
    float* hp = h + (size_t)d * BATCH * CH;
    const float* xep = xe + (size_t)e * BATCH;
    __shared__ float wv[CH];
    if (threadIdx.x < CH) wv[threadIdx.x] = w1[e * CH + threadIdx.x];
    __syncthreads();
    #pragma unroll
    for (int it = 0; it < (BATCH * CH) / 256; ++it) {
        const int i = it * 256 + threadIdx.x;
        const int b = i >> 4, c = i & 15;
        unsafeAtomicAdd(&hp[i], xep[b] * wv[c]);
    }
}

// ---------------------------------------------------------------------------
// Batchnorm (over batch, per channel) + ELU, in-place on a 128x16 LDS tile.
// ---------------------------------------------------------------------------
__device__ __forceinline__ void bn_elu_tile(
    float* tile, const float* __restrict__ g, const float* __restrict__ be,
    float (*ps)[CH], float (*pq)[CH], float* sc, float* sh, int t)
{
    const int c = t & 15, grp = t >> 4;        // 16 groups x 16 channels
    float s = 0.f, q = 0.f;
    #pragma unroll
    for (int r = grp * 8; r < grp * 8 + 8; ++r) {
        const float v = tile[r * CH + c];
        s += v; q += v * v;
    }
    ps[grp][c] = s; pq[grp][c] = q;
    __syncthreads();
    if (t < CH) {
        float S = 0.f, Q = 0.f;
        #pragma unroll
        for (int gi = 0; gi < 16; ++gi) { S += ps[gi][t]; Q += pq[gi][t]; }
        const float m   = S * (1.0f / BATCH);
        const float var = Q * (1.0f / BATCH) - m * m;
        const float rstd = __frsqrt_rn(var + EPSV);
        const float sl = rstd * g[t];
        sc[t] = sl;
        sh[t] = be[t] - m * sl;
    }
    __syncthreads();
    #pragma unroll
    for (int it = 0; it < (BATCH * CH) / 256; ++it) {
        const int i = it * 256 + t;
        const int cc = i & 15;
        const float v = tile[i] * sc[cc] + sh[cc];
        tile[i] = v > 0.f ? v : (__expf(v) - 1.0f);   // ELU(alpha=1)
    }
    __syncthreads();
}

// ---------------------------------------------------------------------------
// Fused per-node: BN1 + ELU -> (128x16)@(16x16) fp32 WMMA -> +b2 -> BN2+ELU
// One 256-thread block (8 wave32) per node; wave w owns rows w*16..w*16+15.
// Tile fill uses CDNA5 async global->LDS (ASYNCcnt) to skip the VGPR bounce.
// ---------------------------------------------------------------------------
__global__ __launch_bounds__(256) void k_node(
    const float* __restrict__ h,     // [N][B][C]
    const float* __restrict__ w2,    // [N][C][C]
    const float* __restrict__ b2,    // [N][C]
    const float* __restrict__ g1, const float* __restrict__ be1,
    const float* __restrict__ g2, const float* __restrict__ be2,
    const unsigned char* __restrict__ fmask,
    float* __restrict__ hout)        // [N][B][C]
{
    __shared__ float tile[BATCH * CH];   // 8 KB node tile
    __shared__ float wt[CH * CH];        // w2m[n]
    __shared__ float ps[16][CH];
    __shared__ float pq[16][CH];
    __shared__ float sc[CH];
    __shared__ float sh[CH];

    const int n = blockIdx.x;
    const int t = threadIdx.x;

    // --- async global->LDS tile fill (2 x b128 per thread = 8 KB total) ---
    // LDS byte address = low 32 bits of the flat shared pointer (ISA 10.2).
    const float* hp = h + (size_t)n * BATCH * CH;
    {
        const unsigned ldsAddr = (unsigned)(size_t)(&tile[0]) + (unsigned)t * 16u;
        const unsigned gOff    = (unsigned)t * 16u;    // per-lane byte offset
        asm volatile(
            "global_load_async_to_lds_b128 %0, %1, %2\n\t"
            "global_load_async_to_lds_b128 %0, %1, %2 offset:4096"
            :
            : "v"(ldsAddr), "v"(gOff), "s"(hp)
            : "memory");
    }
    if (t < CH * CH) {
        const float fm = fmask[n] ? 1.0f : 0.0f;
        wt[t] = w2[(size_t)n * CH * CH + t] * fm;
    }
    asm volatile("s_wait_asynccnt 0" ::: "memory");
    __syncthreads();

    // BN1 + ELU (tile-local: BN is over batch axis only)
    bn_elu_tile(tile, g1 + n * CH, be1 + n * CH, ps, pq, sc, sh, t);

    // WMMA: D(16x16) = A(16x16-rows of batch) x W(16x16), K split in 4 steps
    const int wave  = t >> 5;
    const int lane  = t & 31;
    const int half  = lane >> 4;     // 0: lanes 0-15, 1: lanes 16-31
    const int idx16 = lane & 15;     // A row (M) / B,D col (N)
    const int kh    = half * 2;      // K sub-offset within a 4-wide step

    v8f acc = {0.f, 0.f, 0.f, 0.f, 0.f, 0.f, 0.f, 0.f};
    #pragma unroll
    for (int k0 = 0; k0 < CH; k0 += 4) {
        v2f a, bb;
        const float* arow = &tile[(wave * 16 + idx16) * CH + k0 + kh];
        a.x  = arow[0];
        a.y  = arow[1];
        bb.x = wt[(k0 + kh) * CH + idx16];
        bb.y = wt[(k0 + kh + 1) * CH + idx16];
        acc = __builtin_amdgcn_wmma_f32_16x16x4_f32(
                  false, a, false, bb, (short)0, acc, false, false);
    }
    const float b2v = b2[n * CH + idx16];
    __syncthreads();                 // everyone done reading tile
    const int rb = half * 8;         // D layout: VGPR r -> rows r / r+8
    #pragma unroll
    for (int r = 0; r < 8; ++r)
        tile[(wave * 16 + rb + r) * CH + idx16] = acc[r] + b2v;
    __syncthreads();

    // BN2 + ELU
    bn_elu_tile(tile, g2 + n * CH, be2 + n * CH, ps, pq, sc, sh, t);

    float* op = hout + (size_t)n * BATCH * CH;
    #pragma unroll
    for (int it = 0; it < (BATCH * CH) / 256; ++it)
        op[it * 256 + t] = tile[it * 256 + t];
}

// ---------------------------------------------------------------------------
// Edge update: xe[e][b] = sum_c hout[src[e]][b][c]*w3m[e][c] + b3[e] + x0[e][b]
// ---------------------------------------------------------------------------
__global__ __launch_bounds__(128) void k_edge(
    const float* __restrict__ hout, const int* __restrict__ ei,
    const float* __restrict__ w3m, const float* __restrict__ b3,
    const float* __restrict__ x0, float* __restrict__ xe)
{
    const int e = blockIdx.x;
    const int b = threadIdx.x;
    __shared__ float wv[CH];
    const int s = ei[e];
    if (b < CH) wv[b] = w3m[e * CH + b];
    __syncthreads();
    const float* hp = hout + ((size_t)s * BATCH + b) * CH;
    float acc = b3[e];
    #pragma unroll
    for (int c = 0; c < CH; ++c) acc += hp[c] * wv[c];
    xe[(size_t)e * BATCH + b] = acc + x0[(size_t)e * BATCH + b];
}

// ---------------------------------------------------------------------------
// Final reduction: tmp[n][b] = sum_{e:dst=n} xe[e][b]; out[b][n] = tmp*mask[n]
// ---------------------------------------------------------------------------
__global__ __launch_bounds__(256) void k_zero(float* __restrict__ p, int count)
{
    const int i = blockIdx.x * 256 + threadIdx.x;
    if (i < count) p[i] = 0.0f;
}

__global__ __launch_bounds__(128) void k_scatter_out(
    const float* __restrict__ xe, const int* __restrict__ ei,
    float* __restrict__ tmp)
{
    const int e = blockIdx.x;
    const int b = threadIdx.x;
    const int d = ei[NE + e];
    unsafeAtomicAdd(&tmp[(size_t)d * BATCH + b], xe[(size_t)e * BATCH + b]);
}

__global__ __launch_bounds__(256) void k_out(
    const float* __restrict__ tmp, const unsigned char* __restrict__ omask,
    float* __restrict__ out)
{
    const int i = blockIdx.x * 256 + threadIdx.x;   // i = b*N + n
    if (i < BATCH * NN) {
        const int n = i % NN;
        const int b = i / NN;
        out[i] = tmp[(size_t)n * BATCH + b] * (omask[n] ? 1.0f : 0.0f);
    }
}

// ---------------------------------------------------------------------------
extern "C" void kernel_launch(void* const* d_in, const int* in_sizes, int n_in,
                              void* d_out, int out_size, void* d_ws, size_t ws_size,
                              hipStream_t stream)
{
    const float* x   = (const float*)d_in[0];
    const float* w1  = (const float*)d_in[1];
    const float* b1  = (const float*)d_in[2];
    const float* w2  = (const float*)d_in[3];
    const float* b2  = (const float*)d_in[4];
    const float* w3  = (const float*)d_in[5];
    const float* b3  = (const float*)d_in[6];
    const float* g1  = (const float*)d_in[7];
    const float* be1 = (const float*)d_in[8];
    const float* g2  = (const float*)d_in[9];
    const float* be2 = (const float*)d_in[10];
    const int*   ei  = (const int*)d_in[11];
    const unsigned char* fm = (const unsigned char*)d_in[12];
    const unsigned char* om = (const unsigned char*)d_in[13];
    float* out = (float*)d_out;

    // Workspace carve-up (fp32): ~56 MB total
    float* ws   = (float*)d_ws;
    float* x0   = ws;                          // E*B
    float* xe   = x0   + (size_t)NE * BATCH;   // E*B
    float* h    = xe   + (size_t)NE * BATCH;   // N*B*C
    float* hout = h    + (size_t)NN * BATCH * CH; // N*B*C
    float* w3m  = hout + (size_t)NN * BATCH * CH; // E*C
    float* tmp  = w3m  + (size_t)NE * CH;      // N*B

    k_seed<<<NE, 128, 0, stream>>>(x, ei, w3, fm, x0, xe, w3m);

    const int hElems = NN * BATCH * CH;
    for (int layer = 0; layer < LAYERS; ++layer) {
        k_init_h<<<(hElems + 255) / 256, 256, 0, stream>>>(b1, h);
        k_scatter<<<NE, 256, 0, stream>>>(xe, ei, w1, h);
        k_node<<<NN, 256, 0, stream>>>(h, w2, b2, g1, be1, g2, be2, fm, hout);
        k_edge<<<NE, 128, 0, stream>>>(hout, ei, w3m, b3, x0, xe);
    }

    k_zero<<<(NN * BATCH + 255) / 256, 256, 0, stream>>>(tmp, NN * BATCH);
    k_scatter_out<<<NE, 128, 0, stream>>>(xe, ei, tmp);
    k_out<<<(BATCH * NN + 255) / 256, 256, 0, stream>>>(tmp, om, out);
}